// Model_85968065396889
// MI455X (gfx1250) — compile-verified
//
#include <hip/hip_runtime.h>
#include <cstdint>

// ---------------------------------------------------------------------------
// EmbeddingBag-mean + row gather for MI455X (gfx1250).
//
// out[b, :] = mean over tokens t with seg_ids[t] == tw_graph_idx[b] of
//             emb[flat_idx[t], :]        (zeros if the segment is empty)
//
// Roofline: ~600MB of row gathers served from the L2-resident 60MB table +
// 60MB of write-once output; zero FLOP reuse -> pure data movement.  The
// CDNA5 path that rewards this pattern is the Tensor Data Mover in GATHER
// mode: one TENSOR_LOAD_TO_LDS descriptor DMAs up to 16 embedding rows
// (16-bit row indices, V=50000 < 2^16) into LDS, tracked by TENSORcnt.
// SALU builds descriptors, TDM moves data, VALU only reduces from LDS.
// Output uses non-temporal stores so the table stays L2-resident.
// ---------------------------------------------------------------------------

typedef __attribute__((ext_vector_type(4))) unsigned int u32x4;
typedef __attribute__((ext_vector_type(8))) unsigned int u32x8;
typedef __attribute__((ext_vector_type(4))) float        f32x4;

#define BLOCK 96     // 3 wave32
#define CHUNK 16     // rows per TDM gather descriptor (16-bit index mode max)

__device__ __forceinline__ int lower_bound_i32(const int* __restrict__ a,
                                               int n, int key) {
  int lo = 0, hi = n;
  while (lo < hi) {
    int mid = (lo + hi) >> 1;
    if (a[mid] < key) lo = mid + 1; else hi = mid;
  }
  return lo;
}

__device__ __forceinline__ unsigned rfl(unsigned v) {
  return (unsigned)__builtin_amdgcn_readfirstlane((int)v);
}

__global__ void __launch_bounds__(BLOCK)
embbag_mean_tdm(const float* __restrict__ emb,       // [V, D]
                const int*   __restrict__ flat_idx,  // [T]  token -> vocab row
                const int*   __restrict__ seg_ids,   // [T]  sorted token -> segment
                const int*   __restrict__ gat_idx,   // [B]  output row -> segment
                float*       __restrict__ out,       // [B, D]
                int V, int D, int T) {
  extern __shared__ float stage[];                   // CHUNK * D floats
  const int b   = (int)blockIdx.x;
  const int tid = (int)threadIdx.x;

  // Token range of this output row's segment (uniform across the block;
  // the compiler lowers these to scalar s_load binary searches).
  const int n     = gat_idx[b];
  const int lo    = lower_bound_i32(seg_ids, T, n);
  const int hi    = lower_bound_i32(seg_ids, T, n + 1);
  const int count = hi - lo;

  const int D4  = D >> 2;          // float4 chunks per row (300 -> 75)
  const int rem = D - (D4 << 2);   // scalar tail (0 for D=300)
  const int c0  = tid;
  const int c1  = tid + BLOCK;

  f32x4 acc0 = (f32x4){0.f, 0.f, 0.f, 0.f};
  f32x4 acc1 = (f32x4){0.f, 0.f, 0.f, 0.f};
  float accr = 0.f;

  // ---- Tensor DMA descriptor, group 0 (invariant across chunks) ----------
  const unsigned long long ga = (unsigned long long)(uintptr_t)emb;
  const unsigned lds_byte = (unsigned)(uintptr_t)(&stage[0]);

  const unsigned g0w0 = rfl(1u                  // count = 1 valid descriptor
                            /* bit30 = 0: 16-bit gather indices */
                            | (1u << 31));      // gather_mode = 1
  const unsigned g0w1 = rfl(lds_byte);                          // lds_addr
  const unsigned g0w2 = rfl((unsigned)(ga & 0xFFFFFFFFull));    // global_addr lo
  const unsigned g0w3 = rfl((unsigned)((ga >> 32) & 0x01FFFFFFull)
                            | (2u << 30));      // type = 2 ("image")
  const u32x4 g0 = {g0w0, g0w1, g0w2, g0w3};

  // ---- group 1 words that don't change per chunk -------------------------
  const unsigned uD = (unsigned)D, uV = (unsigned)V;
  const unsigned g1w0 = rfl(2u << 16);                            // data_size = 4B
  const unsigned g1w1 = rfl((uD & 0xFFFFu) << 16);                // tensor_dim0 lo16
  const unsigned g1w2 = rfl((uD >> 16) | ((uV & 0xFFFFu) << 16)); // dim0 hi | dim1 lo
  const unsigned g1w3 = rfl((uV >> 16) | ((uD & 0xFFFFu) << 16)); // dim1 hi | tile_dim0
  const unsigned g1w5 = rfl(uD);                                  // tensor_dim0_stride
  // (workgroup_mask must be 0: not in a cluster)

  for (int pos = lo; pos < hi; pos += CHUNK) {
    const int cnt = (hi - pos < CHUNK) ? (hi - pos) : CHUNK;

    // Pack up to 16 gather row-indices (16-bit) into D# groups 2 and 3.
    unsigned w[8];
#pragma unroll
    for (int j = 0; j < 8; ++j) {
      unsigned i0 = (2 * j     < cnt) ? (unsigned)flat_idx[pos + 2 * j]     : 0u;
      unsigned i1 = (2 * j + 1 < cnt) ? (unsigned)flat_idx[pos + 2 * j + 1] : 0u;
      w[j] = rfl((i0 & 0xFFFFu) | (i1 << 16));
    }

    const u32x8 g1 = {g1w0, g1w1, g1w2, g1w3,
                      rfl((unsigned)cnt),      // tile_dim1 = #valid indices
                      g1w5, 0u, 0u};
    const u32x4 g2 = {w[0], w[1], w[2], w[3]};
    const u32x4 g3 = {w[4], w[5], w[6], w[7]};

    if (tid < 32) {
      // One DMA per chunk, issued by wave 0 only (EXEC is ignored by TDM;
      // issuing from every wave would triplicate the transfer).
      asm volatile("tensor_load_to_lds %0, %1, %2, %3"
                   :: "s"(g0), "s"(g1), "s"(g2), "s"(g3)
                   : "memory");
      __builtin_amdgcn_s_wait_tensorcnt(0);
    }
    __syncthreads();   // staged rows visible to all 3 waves

    // Reduce the staged rows: thread tid owns float4 chunk(s) of the row.
    for (int r = 0; r < cnt; ++r) {
      const float* row = &stage[r * D];
      if (c0 < D4) acc0 += ((const f32x4*)row)[c0];
      if (c1 < D4) acc1 += ((const f32x4*)row)[c1];
      if (tid < rem) accr += row[(D4 << 2) + tid];
    }
    __syncthreads();   // reads done before next DMA overwrites the buffer
  }

  // Write-once result: non-temporal stores keep the table resident in L2.
  const float inv = 1.0f / fmaxf((float)count, 1.0f);
  float* orow = out + (size_t)b * (size_t)D;
  if (c0 < D4) __builtin_nontemporal_store(acc0 * inv, (f32x4*)orow + c0);
  if (c1 < D4) __builtin_nontemporal_store(acc1 * inv, (f32x4*)orow + c1);
  if (tid < rem) __builtin_nontemporal_store(accr * inv, orow + (D4 << 2) + tid);
}

extern "C" void kernel_launch(void* const* d_in, const int* in_sizes, int n_in,
                              void* d_out, int out_size, void* d_ws, size_t ws_size,
                              hipStream_t stream) {
  const float* emb      = (const float*)d_in[0];  // [V, D] f32
  const int*   flat_idx = (const int*)  d_in[1];  // [T]
  const int*   seg_ids  = (const int*)  d_in[2];  // [T] sorted
  const int*   gat_idx  = (const int*)  d_in[3];  // [B]
  float*       out      = (float*)d_out;          // [B, D] f32

  const int T = in_sizes[1];
  const int B = in_sizes[3];
  const int D = out_size / B;        // 300
  const int V = in_sizes[0] / D;     // 50000

  const size_t shmem = (size_t)CHUNK * (size_t)D * sizeof(float);
  hipLaunchKernelGGL(embbag_mean_tdm, dim3((unsigned)B), dim3(BLOCK), shmem,
                     stream, emb, flat_idx, seg_ids, gat_idx, out, V, D, T);
}